// GraphTransformer_87393994539669
// MI455X (gfx1250) — compile-verified
//
#include <hip/hip_runtime.h>
#include <math.h>

#define Nn 20000
#define Ee 100000
#define Rr 4
#define Hh 4
#define Dd 64
#define Ff 128
#define Ll 2
#define DH 256   // H*D
#define EPSf 1e-5f
#define PAD 4    // LDS row padding (floats): keeps banks conflict-free + 16B alignment

typedef float v2f __attribute__((ext_vector_type(2)));
typedef float v8f __attribute__((ext_vector_type(8)));

// ---- ordered-float encoding for atomic segment-max --------------------------
__device__ __forceinline__ unsigned f2ord(float f) {
    unsigned u = __float_as_uint(f);
    return (u & 0x80000000u) ? ~u : (u | 0x80000000u);
}
__device__ __forceinline__ float ord2f(unsigned u) {
    u = (u & 0x80000000u) ? (u & 0x7fffffffu) : ~u;
    return __uint_as_float(u);
}
#define ORD_NEG_INF 0x007FFFFFu   // f2ord(-inf)

// ---- async global->LDS staging (gfx1250 GLOBAL_LOAD_ASYNC_TO_LDS_B128) ------
#if defined(__has_builtin)
#  if __has_builtin(__builtin_amdgcn_global_load_async_to_lds_b128)
#    define HAVE_ASYNC_LDS 1
#  endif
#endif

#ifdef HAVE_ASYNC_LDS
typedef int v4i __attribute__((vector_size(16)));
typedef __attribute__((address_space(1))) v4i* gv4p;   // global (__device__) ptr
typedef __attribute__((address_space(3))) v4i* lv4p;   // LDS (__shared__) ptr
__device__ __forceinline__ void async_copy16(const float* g, float* l) {
    __builtin_amdgcn_global_load_async_to_lds_b128((gv4p)g, (lv4p)l, 0, 0);
}
__device__ __forceinline__ void async_wait0() {
#  if __has_builtin(__builtin_amdgcn_s_wait_asynccnt)
    __builtin_amdgcn_s_wait_asynccnt(0);
#  else
    asm volatile("s_wait_asynccnt 0" ::: "memory");
#  endif
}
#endif

// Stage a ROWS x COLS f32 tile (row stride ldg in global) into padded LDS.
template <int ROWS, int COLS, int NT>
__device__ __forceinline__ void stage_tile(const float* __restrict__ gbase, int ldg,
                                           float* __restrict__ lds)
{
    const int tid = threadIdx.x;
#ifdef HAVE_ASYNC_LDS
    constexpr int C4 = COLS / 4;
    for (int i = tid; i < ROWS * C4; i += NT) {
        int rr = i / C4, c4 = i % C4;
        async_copy16(gbase + (size_t)rr * ldg + c4 * 4, lds + rr * (COLS + PAD) + c4 * 4);
    }
    async_wait0();
#else
    for (int i = tid; i < ROWS * COLS; i += NT) {
        int rr = i / COLS, cc = i % COLS;
        lds[rr * (COLS + PAD) + cc] = gbase[(size_t)rr * ldg + cc];
    }
#endif
    __syncthreads();
}

// ---- wave-level 32x16 f32 WMMA tile: two accumulators share one B fragment --
// A layout (V_WMMA_F32_16X16X4_F32): lanes 0-15 -> M=lane; VGPR0=K{0|2}, VGPR1=K{1|3}.
template <int KLEN>
__device__ __forceinline__ void wmma_gemm_tile2(const float* __restrict__ ldsA, int ldaPad,
                                                const float* __restrict__ B, int ldb,
                                                int n0, v8f& acc0, v8f& acc1)
{
    const int lane = threadIdx.x & 31;
    const int mr   = lane & 15;
    const int kh   = (lane >> 4) << 1;      // 0 or 2
    const int col  = n0 + mr;
#pragma unroll
    for (int k = 0; k < KLEN; k += 4) {
        v2f a0, a1, b;
        a0.x = ldsA[mr * ldaPad + k + kh];
        a0.y = ldsA[mr * ldaPad + k + kh + 1];
        a1.x = ldsA[(mr + 16) * ldaPad + k + kh];
        a1.y = ldsA[(mr + 16) * ldaPad + k + kh + 1];
        b.x  = B[(k + kh) * ldb + col];
        b.y  = B[(k + kh + 1) * ldb + col];
        acc0 = __builtin_amdgcn_wmma_f32_16x16x4_f32(false, a0, false, b,
                                                     (short)0, acc0, false, false);
        acc1 = __builtin_amdgcn_wmma_f32_16x16x4_f32(false, a1, false, b,
                                                     (short)0, acc1, false, false);
    }
}

// ---- encoder: x = nf @ Wf + bf + [bet,clo] @ Wc + bc ------------------------
__global__ void enc_kernel(const float* __restrict__ nf, const float* __restrict__ Wf,
                           const float* __restrict__ bf, const float* __restrict__ bet,
                           const float* __restrict__ clo, const float* __restrict__ Wc,
                           const float* __restrict__ bc, float* __restrict__ x)
{
    __shared__ float ldsA[32 * (Ff + PAD)];
    const int m0 = blockIdx.x * 32;
    stage_tile<32, Ff, 128>(nf + (size_t)m0 * Ff, Ff, ldsA);

    const int wave = threadIdx.x >> 5, lane = threadIdx.x & 31;
    const int n0 = wave * 16;                     // 4 waves cover 64 cols
    v8f acc0 = {}, acc1 = {};
    wmma_gemm_tile2<Ff>(ldsA, Ff + PAD, Wf, Dd, n0, acc0, acc1);

    const int col = n0 + (lane & 15);
    const int rb0 = m0 + ((lane >> 4) << 3);
    const float bfc = bf[col] + bc[col];
    const float wc0 = Wc[col], wc1 = Wc[Dd + col];
#pragma unroll
    for (int v = 0; v < 8; ++v) {
        int r0 = rb0 + v, r1 = r0 + 16;
        x[r0 * Dd + col] = acc0[v] + bfc + bet[r0] * wc0 + clo[r0] * wc1;
        x[r1 * Dd + col] = acc1[v] + bfc + bet[r1] * wc0 + clo[r1] * wc1;
    }
}

// ---- QKV projections: {Q,K,V} = x @ W* + b* --------------------------------
__global__ void qkv_kernel(const float* __restrict__ x,
                           const float* __restrict__ Wq, const float* __restrict__ bq,
                           const float* __restrict__ Wk, const float* __restrict__ bk,
                           const float* __restrict__ Wv, const float* __restrict__ bv,
                           float* __restrict__ Q, float* __restrict__ K, float* __restrict__ V)
{
    __shared__ float ldsA[32 * (Dd + PAD)];
    const int m0 = blockIdx.x * 32;
    stage_tile<32, Dd, 256>(x + (size_t)m0 * Dd, Dd, ldsA);

    const int wave = threadIdx.x >> 5, lane = threadIdx.x & 31;
    const int ct = blockIdx.y * 8 + wave;         // 0..47 (3 mats * 16 col tiles)
    const int mat = ct >> 4;
    const int n0 = (ct & 15) * 16;
    const float* W  = (mat == 0) ? Wq : (mat == 1) ? Wk : Wv;
    const float* bb = (mat == 0) ? bq : (mat == 1) ? bk : bv;
    float* Ob       = (mat == 0) ? Q  : (mat == 1) ? K  : V;

    v8f acc0 = {}, acc1 = {};
    wmma_gemm_tile2<Dd>(ldsA, Dd + PAD, W, DH, n0, acc0, acc1);

    const int col = n0 + (lane & 15);
    const int rb0 = m0 + ((lane >> 4) << 3);
    const float bcol = bb[col];
#pragma unroll
    for (int v = 0; v < 8; ++v) {
        Ob[(size_t)(rb0 + v) * DH + col]      = acc0[v] + bcol;
        Ob[(size_t)(rb0 + v + 16) * DH + col] = acc1[v] + bcol;
    }
}

// ---- per-relation state init ------------------------------------------------
__global__ void init_rel_kernel(float* __restrict__ outr, unsigned* __restrict__ mx,
                                float* __restrict__ den)
{
    const int stride = gridDim.x * blockDim.x;
    int i = blockIdx.x * blockDim.x + threadIdx.x;
    for (int j = i; j < Nn * DH; j += stride) outr[j] = 0.f;
    for (int j = i; j < Nn * Hh; j += stride) { mx[j] = ORD_NEG_INF; den[j] = 0.f; }
}

__global__ void init_attn_out_kernel(float* __restrict__ attn_out, const float* __restrict__ bp)
{
    int i = blockIdx.x * blockDim.x + threadIdx.x;
    if (i < Nn * Dd) attn_out[i] = bp[i & (Dd - 1)];
}

// ---- edge scores + segment max ---------------------------------------------
__global__ void scores_kernel(const float* __restrict__ Q, const float* __restrict__ Kb,
                              const int* __restrict__ src, const int* __restrict__ tgt,
                              const float* __restrict__ sinf, const float* __restrict__ sw,
                              float* __restrict__ scb, unsigned* __restrict__ mx)
{
    int idx = blockIdx.x * blockDim.x + threadIdx.x;
    if (idx >= Ee * Hh) return;
    const int e = idx >> 2, h = idx & 3;
    const int s = src[e], t = tgt[e];
    const float4* q = (const float4*)(Q + (size_t)s * DH + h * Dd);
    const float4* k = (const float4*)(Kb + (size_t)t * DH + h * Dd);
    float acc = 0.f;
#pragma unroll
    for (int i = 0; i < 16; ++i) {
        float4 qa = q[i], ka = k[i];
        acc += qa.x * ka.x + qa.y * ka.y + qa.z * ka.z + qa.w * ka.w;
    }
    const float scv = acc * 0.125f * sinf[s] * sw[h * Rr];   // sw strided by R
    scb[idx] = scv;
    atomicMax(&mx[s * Hh + h], f2ord(scv));
}

// ---- exp + segment sum ------------------------------------------------------
__global__ void expsum_kernel(const int* __restrict__ src, float* __restrict__ scb,
                              const unsigned* __restrict__ mx, float* __restrict__ den)
{
    int idx = blockIdx.x * blockDim.x + threadIdx.x;
    if (idx >= Ee * Hh) return;
    const int e = idx >> 2, h = idx & 3;
    const int s = src[e];
    const float m = ord2f(mx[s * Hh + h]);
    const float v = __expf(scb[idx] - m);
    scb[idx] = v;
    atomicAdd(&den[s * Hh + h], v);
}

// ---- normalized scatter: out[src] += attn * V[tgt] -------------------------
__global__ void scatter_kernel(const int* __restrict__ src, const int* __restrict__ tgt,
                               const float* __restrict__ V, const float* __restrict__ scb,
                               const float* __restrict__ den, float* __restrict__ outr)
{
    int idx = blockIdx.x * blockDim.x + threadIdx.x;
    if (idx >= Ee * Hh * 16) return;
    const int d4 = idx & 15;
    const int h  = (idx >> 4) & 3;
    const int e  = idx >> 6;
    const int s = src[e], t = tgt[e];
    const float a = scb[e * Hh + h] / (den[s * Hh + h] + 1e-10f);
    const float4 vv = *(const float4*)(V + (size_t)t * DH + h * Dd + d4 * 4);
    float* o = outr + (size_t)s * DH + h * Dd + d4 * 4;
    atomicAdd(o + 0, vv.x * a);
    atomicAdd(o + 1, vv.y * a);
    atomicAdd(o + 2, vv.z * a);
    atomicAdd(o + 3, vv.w * a);
}

// ---- projection accumulate: attn_out += outr @ Wp_r ------------------------
__global__ void proj_kernel(const float* __restrict__ outr, const float* __restrict__ Wp_r,
                            float* __restrict__ attn_out)
{
    __shared__ float ldsA[32 * (DH + PAD)];
    const int m0 = blockIdx.x * 32;
    stage_tile<32, DH, 128>(outr + (size_t)m0 * DH, DH, ldsA);

    const int wave = threadIdx.x >> 5, lane = threadIdx.x & 31;
    const int n0 = wave * 16;                     // 4 waves cover 64 cols
    const int col = n0 + (lane & 15);
    const int rb0 = m0 + ((lane >> 4) << 3);
    v8f acc0, acc1;
#pragma unroll
    for (int v = 0; v < 8; ++v) {
        acc0[v] = attn_out[(rb0 + v) * Dd + col];
        acc1[v] = attn_out[(rb0 + v + 16) * Dd + col];
    }
    wmma_gemm_tile2<DH>(ldsA, DH + PAD, Wp_r, Dd, n0, acc0, acc1);
#pragma unroll
    for (int v = 0; v < 8; ++v) {
        attn_out[(rb0 + v) * Dd + col]      = acc0[v];
        attn_out[(rb0 + v + 16) * Dd + col] = acc1[v];
    }
}

// ---- residual + LayerNorm (wave32 shuffle reduction, one wave per node) ----
__global__ void ln_kernel(const float* __restrict__ x, const float* __restrict__ attn_out,
                          const float* __restrict__ g, const float* __restrict__ b,
                          float* __restrict__ dst)
{
    const int wave = threadIdx.x >> 5, lane = threadIdx.x & 31;
    const int nid = blockIdx.x * 8 + wave;
    if (nid >= Nn) return;
    const float* xr = x + (size_t)nid * Dd;
    const float* ar = attn_out + (size_t)nid * Dd;
    const float y0 = xr[lane] + ar[lane];
    const float y1 = xr[lane + 32] + ar[lane + 32];
    float ssum = y0 + y1, ssq = y0 * y0 + y1 * y1;
#pragma unroll
    for (int off = 16; off; off >>= 1) {
        ssum += __shfl_xor(ssum, off, 32);
        ssq  += __shfl_xor(ssq, off, 32);
    }
    const float mean = ssum * (1.f / 64.f);
    const float var  = ssq * (1.f / 64.f) - mean * mean;
    const float rstd = rsqrtf(var + EPSf);
    float* dr = dst + (size_t)nid * Dd;
    dr[lane]      = (y0 - mean) * rstd * g[lane]      + b[lane];
    dr[lane + 32] = (y1 - mean) * rstd * g[lane + 32] + b[lane + 32];
}

// ---------------------------------------------------------------------------
extern "C" void kernel_launch(void* const* d_in, const int* in_sizes, int n_in,
                              void* d_out, int out_size, void* d_ws, size_t ws_size,
                              hipStream_t stream)
{
    const float* nf    = (const float*)d_in[1];
    const float* bet   = (const float*)d_in[2];
    const float* clo   = (const float*)d_in[3];
    const float* sinf  = (const float*)d_in[4];
    const int*   esrc  = (const int*)d_in[5];
    const int*   etgt  = (const int*)d_in[6];
    const float* encWf = (const float*)d_in[7];
    const float* encbf = (const float*)d_in[8];
    const float* encWc = (const float*)d_in[9];
    const float* encbc = (const float*)d_in[10];
    const float* Wq = (const float*)d_in[11];
    const float* bq = (const float*)d_in[12];
    const float* Wk = (const float*)d_in[13];
    const float* bk = (const float*)d_in[14];
    const float* Wv = (const float*)d_in[15];
    const float* bv = (const float*)d_in[16];
    const float* Wp = (const float*)d_in[17];
    const float* bp = (const float*)d_in[18];
    const float* sw = (const float*)d_in[19];
    const float* lng = (const float*)d_in[20];
    const float* lnb = (const float*)d_in[21];

    float* ws = (float*)d_ws;
    float* x        = ws;  ws += Nn * Dd;
    float* Qb       = ws;  ws += Nn * DH;
    float* Kb       = ws;  ws += Nn * DH;
    float* Vb       = ws;  ws += Nn * DH;
    float* outr     = ws;  ws += Nn * DH;
    float* attn_out = ws;  ws += Nn * Dd;
    float* scb      = ws;  ws += Ee * Hh;
    unsigned* mx    = (unsigned*)ws;  ws += Nn * Hh;
    float* den      = ws;  ws += Nn * Hh;

    enc_kernel<<<dim3(Nn / 32), 128, 0, stream>>>(nf, encWf, encbf, bet, clo, encWc, encbc, x);

    const int EH = Ee * Hh;
    for (int l = 0; l < Ll; ++l) {
        init_attn_out_kernel<<<(Nn * Dd + 255) / 256, 256, 0, stream>>>(attn_out, bp + l * Dd);
        for (int r = 0; r < Rr; ++r) {
            const size_t wo = (size_t)(l * Rr + r) * Dd * DH;
            const size_t bo = (size_t)(l * Rr + r) * DH;
            qkv_kernel<<<dim3(Nn / 32, 6), 256, 0, stream>>>(x, Wq + wo, bq + bo,
                                                             Wk + wo, bk + bo,
                                                             Wv + wo, bv + bo, Qb, Kb, Vb);
            init_rel_kernel<<<1024, 256, 0, stream>>>(outr, mx, den);
            scores_kernel<<<(EH + 255) / 256, 256, 0, stream>>>(Qb, Kb, esrc + r * Ee, etgt + r * Ee,
                                                                sinf, sw + l * Hh * Rr + r, scb, mx);
            expsum_kernel<<<(EH + 255) / 256, 256, 0, stream>>>(esrc + r * Ee, scb, mx, den);
            scatter_kernel<<<(EH * 16 + 255) / 256, 256, 0, stream>>>(esrc + r * Ee, etgt + r * Ee,
                                                                      Vb, scb, den, outr);
            proj_kernel<<<dim3(Nn / 32), 128, 0, stream>>>(outr,
                    Wp + (size_t)l * Rr * DH * Dd + (size_t)r * DH * Dd, attn_out);
        }
        float* dst = (l == Ll - 1) ? (float*)d_out : x;
        ln_kernel<<<dim3((Nn + 7) / 8), 256, 0, stream>>>(x, attn_out, lng + l * Dd, lnb + l * Dd, dst);
    }
}